// NMS_37855841747166
// MI455X (gfx1250) — compile-verified
//
#include <hip/hip_runtime.h>
#include <stdint.h>

#define NBOX 8192
#define NCH  5
#define TPB  1024
#define PER  (NBOX / TPB)   // 8 elements per thread
#define IOU_TH 0.1f

// ---- CDNA5 async global->LDS staging (guarded: fall back to plain copy) ----
#if defined(__gfx1250__) && __has_builtin(__builtin_amdgcn_global_load_async_to_lds_b128)
#define USE_ASYNC_LDS 1
#else
#define USE_ASYNC_LDS 0
#endif

typedef int v4i __attribute__((ext_vector_type(4)));
typedef __attribute__((address_space(1))) v4i* gptr_v4i;   // global (AS1) int4*
typedef __attribute__((address_space(3))) v4i* lptr_v4i;   // LDS    (AS3) int4*

__global__ __launch_bounds__(TPB) void nms_gfx1250_kernel(const float* __restrict__ in,
                                                          float* __restrict__ out)
{
    const int b = blockIdx.x;
    const int t = threadIdx.x;

    extern __shared__ char smem[];
    float*         raw   = (float*)smem;                  // [5][8192]  160 KB
    float*         sS    = raw + NCH * NBOX;              // [8192]      32 KB
    int*           idS   = (int*)(sS + NBOX);             // [8192]      32 KB
    unsigned char* keepS = (unsigned char*)(idS + NBOX);  // [8192]       8 KB
    unsigned char* keepO = keepS + NBOX;                  // [8192]       8 KB
    int*           scanS = (int*)(keepO + NBOX);          // [1024]       4 KB

    const float* gin  = in  + (size_t)b * NCH * NBOX;
    float*       gout = out + (size_t)b * NCH * NBOX;

    // ---------------- Phase A: stage [5,8192] slice into LDS ----------------
    {
        const float4* g4 = (const float4*)gin;
        float4*       l4 = (float4*)raw;
        __builtin_prefetch(gin + t * 16, 0, 0);   // global_prefetch_b8
#if USE_ASYNC_LDS
        #pragma unroll
        for (int q = 0; q < (NCH * NBOX / 4) / TPB; ++q) {
            int v = t + q * TPB;
            __builtin_amdgcn_global_load_async_to_lds_b128(
                (gptr_v4i)(g4 + v),
                (lptr_v4i)(l4 + v),
                0, 0);
        }
#if __has_builtin(__builtin_amdgcn_s_wait_asynccnt)
        __builtin_amdgcn_s_wait_asynccnt(0);
#else
        asm volatile("s_wait_asynccnt 0" ::: "memory");
#endif
#else
        #pragma unroll
        for (int q = 0; q < (NCH * NBOX / 4) / TPB; ++q) {
            int v = t + q * TPB;
            l4[v] = g4[v];
        }
#endif
    }
    __syncthreads();

    // ---------------- Phase B: init (score, idx) pairs ----------------------
    #pragma unroll
    for (int k = 0; k < PER; ++k) {
        int p = t + k * TPB;
        sS[p]  = raw[4 * NBOX + p];
        idS[p] = p;
    }
    __syncthreads();

    // Bitonic sort: descending score, ties broken by ascending original index
    // (strict total order => exactly matches JAX's stable argsort(-scores)).
    for (int k = 2; k <= NBOX; k <<= 1) {
        for (int j = k >> 1; j > 0; j >>= 1) {
            #pragma unroll
            for (int s = 0; s < PER; ++s) {
                int i = t + s * TPB;
                int l = i ^ j;
                if (l > i) {
                    float si = sS[i], sl = sS[l];
                    int   ii = idS[i], il = idS[l];
                    bool before_l = (sl > si) || (sl == si && il < ii);
                    bool asc      = ((i & k) == 0);
                    if (before_l == asc) {
                        sS[i] = sl; sS[l] = si;
                        idS[i] = il; idS[l] = ii;
                    }
                }
            }
            __syncthreads();
        }
    }

    // ---------------- Phase C: keep init + per-thread box registers ---------
    float x1r[PER], y1r[PER], x2r[PER], y2r[PER], ar[PER];
    #pragma unroll
    for (int k = 0; k < PER; ++k) {
        int p = t + k * TPB;
        keepS[p] = (sS[p] > 0.0f) ? 1 : 0;
        int o = idS[p];
        float bx = raw[0 * NBOX + o], by = raw[1 * NBOX + o];
        float bw = raw[2 * NBOX + o], bh = raw[3 * NBOX + o];
        x1r[k] = bx - 0.5f * bw;  y1r[k] = by - 0.5f * bh;
        x2r[k] = bx + 0.5f * bw;  y2r[k] = by + 0.5f * bh;
        ar[k]  = (x2r[k] - x1r[k]) * (y2r[k] - y1r[k]);
    }
    __syncthreads();

    // ---------------- Greedy suppression (sequential over sorted order) -----
    for (int i = 0; i < NBOX; ++i) {
        if (keepS[i]) {
            int o = idS[i];
            float bx = raw[0 * NBOX + o], by = raw[1 * NBOX + o];
            float bw = raw[2 * NBOX + o], bh = raw[3 * NBOX + o];
            float px1 = bx - 0.5f * bw, py1 = by - 0.5f * bh;
            float px2 = bx + 0.5f * bw, py2 = by + 0.5f * bh;
            float pa  = (px2 - px1) * (py2 - py1);
            #pragma unroll
            for (int k = 0; k < PER; ++k) {
                int p = t + k * TPB;
                float iw = fminf(x2r[k], px2) - fmaxf(x1r[k], px1);
                float ih = fminf(y2r[k], py2) - fmaxf(y1r[k], py1);
                iw = fmaxf(iw, 0.0f); ih = fmaxf(ih, 0.0f);
                float inter = iw * ih;
                float uni   = ar[k] + pa - inter;
                float iou   = inter / (uni + 1e-8f);
                if (!(iou < IOU_TH) && p != i) keepS[p] = 0;  // exclusive byte ownership
            }
        }
        __syncthreads();
    }

    // ---------------- Phase D: map keep mask back to original order ---------
    #pragma unroll
    for (int k = 0; k < PER; ++k) {
        int p = t + k * TPB;
        keepO[idS[p]] = keepS[p];
    }
    __syncthreads();

    // Block-wide exclusive scan over per-thread kept counts (Kogge-Stone)
    const int cbase = t * PER;
    int cnt = 0;
    #pragma unroll
    for (int k = 0; k < PER; ++k) cnt += keepO[cbase + k];
    scanS[t] = cnt;
    __syncthreads();
    for (int off = 1; off < TPB; off <<= 1) {
        int v   = scanS[t];
        int add = (t >= off) ? scanS[t - off] : 0;
        __syncthreads();
        scanS[t] = v + add;
        __syncthreads();
    }
    int dest = (t == 0) ? 0 : scanS[t - 1];

    // Zero the whole output slice, then scatter kept columns to the front.
    {
        float4* o4 = (float4*)gout;
        #pragma unroll
        for (int q = 0; q < (NCH * NBOX / 4) / TPB; ++q)
            o4[t + q * TPB] = make_float4(0.f, 0.f, 0.f, 0.f);
    }
    __syncthreads();
    for (int k = 0; k < PER; ++k) {
        int c = cbase + k;
        if (keepO[c]) {
            #pragma unroll
            for (int ch = 0; ch < NCH; ++ch)
                gout[ch * NBOX + dest] = raw[ch * NBOX + c];
            ++dest;
        }
    }
}

extern "C" void kernel_launch(void* const* d_in, const int* in_sizes, int n_in,
                              void* d_out, int out_size, void* d_ws, size_t ws_size,
                              hipStream_t stream) {
    const float* in  = (const float*)d_in[0];
    float*       out = (float*)d_out;
    int B = out_size / (NCH * NBOX);             // = 4
    size_t shmem = (size_t)NCH * NBOX * sizeof(float)   // raw slice  160 KB
                 + (size_t)NBOX * sizeof(float)         // scores      32 KB
                 + (size_t)NBOX * sizeof(int)           // indices     32 KB
                 + (size_t)NBOX                          // keep sorted  8 KB
                 + (size_t)NBOX                          // keep orig    8 KB
                 + (size_t)TPB * sizeof(int);           // scan         4 KB
    nms_gfx1250_kernel<<<dim3(B), dim3(TPB), shmem, stream>>>(in, out);
}